// SessionGraph_59966333387418
// MI455X (gfx1250) — compile-verified
//
#include <hip/hip_runtime.h>

// ---------------- problem constants (from reference) ----------------
constexpr int kB = 512;
constexpr int kL = 100;
constexpr int kE = 50;
constexpr int kD = 512;
constexpr float kAlpha = 0.2f;
constexpr float kNeg = -9.0e15f;

typedef __attribute__((ext_vector_type(2))) float v2f;
typedef __attribute__((ext_vector_type(8))) float v8f;

// ---------------- workspace layout (float offsets) ----------------
constexpr int OFF_C0    = 0;                    // 1 float (padded)
constexpr int OFF_W2A   = 16;                   // 512
constexpr int OFF_W2A2  = 16 + 512;             // 512
constexpr int OFF_W3A2  = 16 + 1024;            // 512
constexpr int OFF_SNODE = 2048;                 // B*L
constexpr int OFF_XA2   = OFF_SNODE + kB * kL;  // B*L
constexpr int OFF_E4A2  = OFF_XA2 + kB * kL;    // B*E

__device__ __forceinline__ float wave_sum(float v) {
  #pragma unroll
  for (int o = 16; o > 0; o >>= 1) v += __shfl_xor(v, o, 32);
  return v;
}
__device__ __forceinline__ float wave_max(float v) {
  #pragma unroll
  for (int o = 16; o > 0; o >>= 1) v = fmaxf(v, __shfl_xor(v, o, 32));
  return v;
}
__device__ __forceinline__ float leaky(float x) {
  return x >= 0.0f ? x : kAlpha * x;
}
__device__ __forceinline__ v8f wmma_f32x4(v2f a, v2f b, v8f c) {
  // D = A(16x4 f32) * B(4x16 f32) + C(16x16 f32): exact fp32 matrix FMA
  return __builtin_amdgcn_wmma_f32_16x16x4_f32(
      /*neg_a=*/false, a, /*neg_b=*/false, b,
      /*c_mod=*/(short)0, c, /*reuse_a=*/false, /*reuse_b=*/false);
}

// ---------------- kernel 0: collapse W2/W3 through the a-vectors ----------------
// w2a = W2 @ a[D:2D], w2a2 = W2 @ a2[0:D], w3a2 = W3 @ a2[D:2D], c0 = ctx . a[0:D]
__global__ void prep_kernel(const float* __restrict__ W2, const float* __restrict__ W3,
                            const float* __restrict__ ctx, const float* __restrict__ a,
                            const float* __restrict__ a2, float* __restrict__ ws) {
  int tid = blockIdx.x * blockDim.x + threadIdx.x;
  if (tid < kD) {
    const float* row = W2 + (long)tid * kD;
    float s = 0.f;
    for (int j = 0; j < kD; ++j) s += row[j] * a[kD + j];
    ws[OFF_W2A + tid] = s;
  } else if (tid < 2 * kD) {
    int i = tid - kD;
    const float* row = W2 + (long)i * kD;
    float s = 0.f;
    for (int j = 0; j < kD; ++j) s += row[j] * a2[j];
    ws[OFF_W2A2 + i] = s;
  } else if (tid < 3 * kD) {
    int i = tid - 2 * kD;
    const float* row = W3 + (long)i * kD;
    float s = 0.f;
    for (int j = 0; j < kD; ++j) s += row[j] * a2[kD + j];
    ws[OFF_W3A2 + i] = s;
  } else if (tid == 3 * kD) {
    float s = 0.f;
    for (int j = 0; j < kD; ++j) s += ctx[j] * a[j];
    ws[OFF_C0] = s;
  }
}

// ---------------- kernel 1: s_node[b,l] and (x4 @ a2_node)[b,l] ----------------
// one wave32 per (b,l) row; emb1 row gathered once (L2-resident)
__global__ void snode_kernel(const int* __restrict__ inputs, const float* __restrict__ emb1,
                             const float* __restrict__ ws, float* __restrict__ snode,
                             float* __restrict__ xa2) {
  int row = blockIdx.x * 8 + (threadIdx.x >> 5);  // b*L + l  (grid is exact)
  int lane = threadIdx.x & 31;
  const float* x = emb1 + (long)inputs[row] * kD;
  const float* w2a = ws + OFF_W2A;
  const float* w2a2 = ws + OFF_W2A2;
  float s1 = 0.f, s2 = 0.f;
  #pragma unroll
  for (int j = 0; j < 4; ++j) {
    int d = j * 128 + lane * 4;
    float4 xv = *(const float4*)(x + d);
    float4 w1 = *(const float4*)(w2a + d);
    float4 w2 = *(const float4*)(w2a2 + d);
    s1 += xv.x * w1.x + xv.y * w1.y + xv.z * w1.z + xv.w * w1.w;
    s2 += xv.x * w2.x + xv.y * w2.y + xv.z * w2.z + xv.w * w2.w;
  }
  s1 = wave_sum(s1);
  s2 = wave_sum(s2);
  if (lane == 0) {
    snode[row] = leaky(ws[OFF_C0] + s1);
    xa2[row] = s2;
  }
}

// ---------------- kernel 2: edge = softmax_L(mask(s_node)) @ x via WMMA ----------------
// grid = (B, D/128); block = 256 (8 wave32). A tile [64x100] in LDS (rows 50..63 zero),
// B fragments gathered straight from emb1 (fits in L2).
__global__ void edge_kernel(const int* __restrict__ inputs, const int* __restrict__ HT,
                            const float* __restrict__ emb1, const float* __restrict__ snode,
                            float* __restrict__ edge) {
  const int b = blockIdx.x;
  const int nbase = blockIdx.y * 128;
  __shared__ float attW[64 * 100];
  __shared__ int sIdx[kL];
  const int tid = threadIdx.x;
  const int wave = tid >> 5, lane = tid & 31;

  for (int i = tid; i < kL; i += 256) sIdx[i] = inputs[b * kL + i];
  for (int i = tid; i < 64 * 100; i += 256) attW[i] = 0.0f;
  __syncthreads();

  // masked softmax over L for each hyperedge e (one wave per e, round-robin)
  for (int e = wave; e < kE; e += 8) {
    float v[4];
    float m = kNeg;
    #pragma unroll
    for (int j = 0; j < 4; ++j) {
      int l = lane + j * 32;
      v[j] = kNeg;
      if (l < kL) {
        float s = snode[b * kL + l];
        v[j] = (HT[((long)b * kE + e) * kL + l] > 0) ? s : kNeg;
        m = fmaxf(m, v[j]);
      }
    }
    m = wave_max(m);
    float w[4];
    float sum = 0.f;
    #pragma unroll
    for (int j = 0; j < 4; ++j) {
      int l = lane + j * 32;
      w[j] = 0.f;
      if (l < kL) { w[j] = __expf(v[j] - m); sum += w[j]; }
    }
    sum = wave_sum(sum);
    float inv = 1.0f / sum;
    #pragma unroll
    for (int j = 0; j < 4; ++j) {
      int l = lane + j * 32;
      if (l < kL) attW[e * 100 + l] = w[j] * inv;
    }
  }
  __syncthreads();

  // GEMM: C[64 x 128] = attW[64 x 100] * x[100 x 128-slice]
  const int mtile = wave & 3;   // 4 M-tiles cover padded E
  const int ngrp = wave >> 2;   // 2 groups of 4 N-tiles
  const int m = lane & 15;
  const int khalf = lane >> 4;  // VGPR K-split per ISA layout
  const int arow = mtile * 16 + m;
  v8f acc0 = {0,0,0,0,0,0,0,0}, acc1 = acc0, acc2 = acc0, acc3 = acc0;
  v8f acc[4];
  acc[0] = acc0; acc[1] = acc1; acc[2] = acc2; acc[3] = acc3;

  for (int ks = 0; ks < 25; ++ks) {           // K = L = 100 = 25 * 4
    const int k0 = ks * 4 + khalf * 2;
    v2f afrag;
    afrag.x = attW[arow * 100 + k0];
    afrag.y = attW[arow * 100 + k0 + 1];
    const float* r0 = emb1 + (long)sIdx[k0] * kD;
    const float* r1 = emb1 + (long)sIdx[k0 + 1] * kD;
    #pragma unroll
    for (int t = 0; t < 4; ++t) {
      int col = nbase + (ngrp * 4 + t) * 16 + m;
      v2f bfrag;
      bfrag.x = r0[col];
      bfrag.y = r1[col];
      acc[t] = wmma_f32x4(afrag, bfrag, acc[t]);
    }
  }
  #pragma unroll
  for (int t = 0; t < 4; ++t) {
    int col = nbase + (ngrp * 4 + t) * 16 + m;
    int rbase = mtile * 16 + khalf * 8;       // C/D layout: lanes>=16 hold M+8
    #pragma unroll
    for (int r = 0; r < 8; ++r) {
      int e = rbase + r;
      if (e < kE) edge[((long)b * kE + e) * kD + col] = acc[t][r];
    }
  }
}

// ---------------- kernel 2b: e4a2[b,e] = edge[b,e,:] . w3a2 ----------------
__global__ void e4a2_kernel(const float* __restrict__ edge, const float* __restrict__ ws,
                            float* __restrict__ e4a2) {
  int row = blockIdx.x * 8 + (threadIdx.x >> 5);  // b*E + e (grid exact)
  int lane = threadIdx.x & 31;
  const float* er = edge + (long)row * kD;
  const float* w = ws + OFF_W3A2;
  float s = 0.f;
  #pragma unroll
  for (int j = 0; j < 4; ++j) {
    int d = j * 128 + lane * 4;
    float4 ev = *(const float4*)(er + d);
    float4 wv = *(const float4*)(w + d);
    s += ev.x * wv.x + ev.y * wv.y + ev.z * wv.z + ev.w * wv.w;
  }
  s = wave_sum(s);
  if (lane == 0) e4a2[row] = s;
}

// ---------------- kernel 3: node = softmax_E(mask(leaky(xa2+e4a2))) @ edge + x ----------------
// grid = (B, D/128); block = 256. A tile [112 x 52] zero-padded in LDS.
__global__ void node_kernel(const int* __restrict__ inputs, const int* __restrict__ HT,
                            const float* __restrict__ emb1, const float* __restrict__ edge,
                            const float* __restrict__ ws, float* __restrict__ out) {
  const int b = blockIdx.x;
  const int nbase = blockIdx.y * 128;
  __shared__ float attN[112 * 52];
  __shared__ int sIdx[kL];
  __shared__ float sE4[kE];
  __shared__ float sX[kL];
  const int tid = threadIdx.x;
  const int wave = tid >> 5, lane = tid & 31;

  for (int i = tid; i < kL; i += 256) {
    sIdx[i] = inputs[b * kL + i];
    sX[i] = ws[OFF_XA2 + b * kL + i];
  }
  for (int i = tid; i < kE; i += 256) sE4[i] = ws[OFF_E4A2 + b * kE + i];
  for (int i = tid; i < 112 * 52; i += 256) attN[i] = 0.0f;
  __syncthreads();

  // per node l: masked softmax over the 50 hyperedges (one wave per l)
  for (int l = wave; l < kL; l += 8) {
    float xv = sX[l];
    float v0 = kNeg, v1 = kNeg;
    int e0 = lane, e1 = lane + 32;
    if (e0 < kE)
      v0 = (HT[((long)b * kE + e0) * kL + l] > 0) ? leaky(xv + sE4[e0]) : kNeg;
    if (e1 < kE)
      v1 = (HT[((long)b * kE + e1) * kL + l] > 0) ? leaky(xv + sE4[e1]) : kNeg;
    float m = wave_max(fmaxf(v0, v1));
    float w0 = (e0 < kE) ? __expf(v0 - m) : 0.f;
    float w1 = (e1 < kE) ? __expf(v1 - m) : 0.f;
    float sum = wave_sum(w0 + w1);
    float inv = 1.0f / sum;
    if (e0 < kE) attN[l * 52 + e0] = w0 * inv;
    if (e1 < kE) attN[l * 52 + e1] = w1 * inv;
  }
  __syncthreads();

  // GEMM: C[112 x 128] = attN[112 x 52] * edge[50 x 128-slice]; each wave owns one N-tile
  const int m = lane & 15;
  const int khalf = lane >> 4;
  const int col = nbase + wave * 16 + m;
  v8f zero = {0,0,0,0,0,0,0,0};
  v8f acc[7];
  #pragma unroll
  for (int t = 0; t < 7; ++t) acc[t] = zero;

  for (int ks = 0; ks < 13; ++ks) {           // K = E = 50 padded to 52
    const int k0 = ks * 4 + khalf * 2;
    v2f bfrag;
    bfrag.x = (k0 < kE)     ? edge[((long)b * kE + k0) * kD + col]     : 0.f;
    bfrag.y = (k0 + 1 < kE) ? edge[((long)b * kE + k0 + 1) * kD + col] : 0.f;
    #pragma unroll
    for (int mt = 0; mt < 7; ++mt) {
      int arow = mt * 16 + m;
      v2f afrag;
      afrag.x = attN[arow * 52 + k0];
      afrag.y = attN[arow * 52 + k0 + 1];
      acc[mt] = wmma_f32x4(afrag, bfrag, acc[mt]);
    }
  }
  const long BLD = (long)kB * kL * kD;
  #pragma unroll
  for (int mt = 0; mt < 7; ++mt) {
    int rbase = mt * 16 + khalf * 8;
    #pragma unroll
    for (int r = 0; r < 8; ++r) {
      int l = rbase + r;
      if (l < kL) {
        float xval = emb1[(long)sIdx[l] * kD + col];   // residual: node += x (L2 hit)
        float val = acc[mt][r] + xval;
        long o = ((long)b * kL + l) * kD + col;
        out[o] = val;            // output 0: node
        out[BLD + o] = val;      // output 1: node (duplicated in tuple)
      }
    }
  }
}

// ---------------- kernel 4: nodes2 = emb2[inputs] (overwrites edge scratch) ----------------
__global__ void gather2_kernel(const int* __restrict__ inputs, const float* __restrict__ emb2,
                               float* __restrict__ out2) {
  long row = blockIdx.x;  // b*L + l
  const float4* src = (const float4*)(emb2 + (long)inputs[row] * kD);
  float4* dst = (float4*)(out2 + row * (long)kD);
  dst[threadIdx.x] = src[threadIdx.x];  // 128 threads x float4 = 512 floats
}

extern "C" void kernel_launch(void* const* d_in, const int* in_sizes, int n_in,
                              void* d_out, int out_size, void* d_ws, size_t ws_size,
                              hipStream_t stream) {
  const int* inputs = (const int*)d_in[0];
  const int* HT = (const int*)d_in[1];
  const float* emb1 = (const float*)d_in[4];
  const float* emb2 = (const float*)d_in[5];
  const float* W2 = (const float*)d_in[6];
  const float* W3 = (const float*)d_in[7];
  const float* ctx = (const float*)d_in[8];
  const float* a = (const float*)d_in[9];
  const float* a2 = (const float*)d_in[10];
  float* out = (float*)d_out;
  float* ws = (float*)d_ws;

  const long BLD = (long)kB * kL * kD;
  // stash edge[B,E,D] (13.1M floats) in output region 2 (26.2M floats);
  // gather2_kernel fully overwrites it at the end -> deterministic.
  float* edge = out + 2 * BLD;

  prep_kernel<<<7, 256, 0, stream>>>(W2, W3, ctx, a, a2, ws);
  snode_kernel<<<(kB * kL) / 8, 256, 0, stream>>>(inputs, emb1, ws,
                                                  ws + OFF_SNODE, ws + OFF_XA2);
  edge_kernel<<<dim3(kB, kD / 128), 256, 0, stream>>>(inputs, HT, emb1,
                                                      ws + OFF_SNODE, edge);
  e4a2_kernel<<<(kB * kE) / 8, 256, 0, stream>>>(edge, ws, ws + OFF_E4A2);
  node_kernel<<<dim3(kB, kD / 128), 256, 0, stream>>>(inputs, HT, emb1, edge, ws, out);
  gather2_kernel<<<kB * kL, 128, 0, stream>>>(inputs, emb2, out + 2 * BLD);
}